// GatedCNNLayer_5437428596749
// MI455X (gfx1250) — compile-verified
//
#include <hip/hip_runtime.h>
#include <stdint.h>

// X: (16, 4096, 1024) f32 | G: (1024, 2) f32 | Gb: (2,) f32 | out: (16, 2047, 1024) f32
#define NB 16
#define NL 4096
#define ND 1024
#define NP 2047
#define WAVES_PER_BLOCK 4
#define NTHREADS (WAVES_PER_BLOCK * 32)
#define ROWS_PER_WAVE 16
#define WAVES_PER_BATCH 128          // ceil(2047/16) = 128 (last wave: 15 valid rows)
#define K_TILE 64
#define N_TILES (ND / K_TILE)        // 16
#define SROW 68                      // LDS row stride (floats): 8B-aligned f2 reads, 0 bank conflicts
#define BUF_FLOATS (ROWS_PER_WAVE * SROW)   // 1088 floats = 4352 B per buffer

typedef float f2  __attribute__((ext_vector_type(2)));
typedef float f4  __attribute__((ext_vector_type(4)));
typedef float v8f __attribute__((ext_vector_type(8)));

__global__ __launch_bounds__(NTHREADS, 1)
void gated_pool_wmma_kernel(const float* __restrict__ X,
                            const float* __restrict__ G,
                            const float* __restrict__ Gb,
                            float* __restrict__ out)
{
    // G interleaved (sG[d*2+g], 8 KB) + per-wave double-buffered mid K-tiles.
    __shared__ __attribute__((aligned(16))) float sG[2 * ND];
    __shared__ __attribute__((aligned(16))) float sMid[WAVES_PER_BLOCK][2][BUF_FLOATS];

    const int lane      = (int)(threadIdx.x & 31);
    const int waveInBlk = (int)(threadIdx.x >> 5);
    const int w         = (int)blockIdx.x * WAVES_PER_BLOCK + waveInBlk;   // 0..2047
    const int b         = w >> 7;            // / WAVES_PER_BATCH
    const int wInB      = w & 127;
    const int p0        = wInB * ROWS_PER_WAVE;

    const int nm   = lane & 15;              // matrix column (mid-row) / A row index
    const int offK = (lane >> 4) << 1;       // K sub-offset: lanes 0-15 -> K{0,1}, 16-31 -> K{2,3}

    // ------------------------------------------------------------------
    // Async-stage G (8 KB = 512 x 16B; 128 threads -> 4 chunks each).
    // ------------------------------------------------------------------
    {
        const uint32_t lbase = (uint32_t)(uintptr_t)&sG[0];
        const uint64_t gbase = (uint64_t)(uintptr_t)G;
#pragma unroll
        for (int i = 0; i < 4; ++i) {
            const int chunk = (int)threadIdx.x + i * NTHREADS;
            uint32_t la = lbase + (uint32_t)chunk * 16u;
            uint64_t ga = gbase + (uint64_t)chunk * 16u;
            asm volatile("global_load_async_to_lds_b128 %0, %1, off"
                         :: "v"(la), "v"(ga) : "memory");
        }
    }

    // Per-lane base addresses for mid-row K-tile staging:
    // instruction i moves rows {2i, 2i+1}; lane -> row (lane>>4)+2i, 16B chunk (lane&15).
    const float* midRow0 = X + ((size_t)b * NL + (size_t)(2 * p0 + 1)) * ND;
    const uint64_t gMid = (uint64_t)(uintptr_t)midRow0
                        + (uint64_t)(lane >> 4) * (uint64_t)(2 * ND * 4)   // next mid row = +8192 B
                        + (uint64_t)(lane & 15) * 16u;
    const uint32_t lMid = (uint32_t)(uintptr_t)&sMid[waveInBlk][0][0]
                        + (uint32_t)(lane >> 4) * (uint32_t)(SROW * 4)
                        + (uint32_t)(lane & 15) * 16u;

    auto stageTile = [&](int tile, int buf) {
        const uint64_t g0 = gMid + (uint64_t)tile * (uint64_t)(K_TILE * 4);
        const uint32_t l0 = lMid + (uint32_t)buf * (uint32_t)(BUF_FLOATS * 4);
#pragma unroll
        for (int i = 0; i < 8; ++i) {
            uint32_t la = l0 + (uint32_t)(i * (2 * SROW * 4));
            uint64_t ga = g0 + (uint64_t)i * (uint64_t)(4 * ND * 4);       // 2 mid rows = 16384 B
            asm volatile("global_load_async_to_lds_b128 %0, %1, off"
                         :: "v"(la), "v"(ga) : "memory");
        }
    };

    // Kick off tile 0 alongside the G stage; one wait covers both.
    stageTile(0, 0);
    asm volatile("s_wait_asynccnt 0x0" ::: "memory");
    __syncthreads();   // sG visible to all waves

    // Prefetch the 17 even rows (phase-2 operands): lane*128B x 32 lanes = one 4KB row per call.
    const float* evenBase = X + ((size_t)b * NL + (size_t)(2 * p0)) * ND;
#pragma unroll
    for (int r = 0; r < 17; ++r)
        __builtin_prefetch((const void*)(evenBase + (size_t)r * (2 * ND) + (size_t)lane * 32), 0, 3);

    // ------------------------------------------------------------------
    // Gate GEMM on the matrix pipe:  D(16x16) += A(16x4) x B(4x16)
    //   A = G^T K-chunk   (rows 0,1 nonzero)         -> from LDS sG
    //   B = 16 mid rows   (lane n owns mid-row n)    -> from LDS tile
    // Result: acc[0] = logit0, acc[1] = logit1 in lanes 0..15 (n = lane).
    // ------------------------------------------------------------------
    v8f acc = {};
    for (int t = 0; t < N_TILES; ++t) {
        if (t < N_TILES - 1) {
            stageTile(t + 1, (t + 1) & 1);
            asm volatile("s_wait_asynccnt 0x8" ::: "memory");   // tile t complete (in-order)
        } else {
            asm volatile("s_wait_asynccnt 0x0" ::: "memory");
        }
        const float* mb = &sMid[waveInBlk][t & 1][0];
#pragma unroll
        for (int s = 0; s < 16; ++s) {
            // B operand: lane n reads mid-row n, local K = s*4+offK (+1). Bank-conflict-free.
            f2 bv = *(const f2*)(mb + nm * SROW + s * 4 + offK);
            // A operand: A[m][k] = G[d0+k, m], nonzero only m < 2.
            const int d0 = t * K_TILE + s * 4 + offK;
            const int gi = (nm < 2) ? (d0 * 2 + nm) : 0;
            const float a0 = sG[gi];
            const float a1 = sG[gi + 2];
            f2 av;
            av.x = (nm < 2) ? a0 : 0.0f;
            av.y = (nm < 2) ? a1 : 0.0f;
            acc = __builtin_amdgcn_wmma_f32_16x16x4_f32(
                      /*neg_a=*/false, av, /*neg_b=*/false, bv,
                      /*c_mod=*/(short)0, acc, /*reuse_a=*/false, /*reuse_b=*/false);
        }
    }

    // Per-lane (lanes 0..15 hold row p0+lane) softmax over the 2 logits.
    const float s0 = acc[0] + Gb[0];
    const float s1 = acc[1] + Gb[1];
    const float mx  = fmaxf(s0, s1);
    const float e0  = __expf(s0 - mx);
    const float e1  = __expf(s1 - mx);
    const float inv = 1.0f / (e0 + e1);
    const float g0v = e0 * inv;
    const float g1v = e1 * inv;

    // ------------------------------------------------------------------
    // Phase 2: out = left*g0 + right*g1, one row at a time.
    // Even rows are read twice across neighboring p -> temporal loads (L2 reuse).
    // Output is write-once -> nontemporal b128 stores.
    // ------------------------------------------------------------------
    float* outBase = out + ((size_t)b * NP + (size_t)p0) * ND;
    for (int m = 0; m < ROWS_PER_WAVE; ++m) {
        if (p0 + m >= NP) break;                       // uniform: only last wave per batch
        const float g0 = __shfl(g0v, m, 32);
        const float g1 = __shfl(g1v, m, 32);
        const f4* Lr = (const f4*)(evenBase + (size_t)m * (2 * ND));
        const f4* Rr = Lr + (2 * ND) / 4;
        f4* Or = (f4*)(outBase + (size_t)m * ND);
#pragma unroll
        for (int c = 0; c < 8; ++c) {
            const int idx = c * 32 + lane;
            f4 lv = Lr[idx];
            f4 rv = Rr[idx];
            f4 o  = lv * g0 + rv * g1;
            __builtin_nontemporal_store(o, &Or[idx]);
        }
    }
}

extern "C" void kernel_launch(void* const* d_in, const int* in_sizes, int n_in,
                              void* d_out, int out_size, void* d_ws, size_t ws_size,
                              hipStream_t stream) {
    const float* X  = (const float*)d_in[0];
    const float* G  = (const float*)d_in[1];
    const float* Gb = (const float*)d_in[2];
    float* out = (float*)d_out;

    const int total_waves = NB * WAVES_PER_BATCH;                 // 2048
    const int blocks = total_waves / WAVES_PER_BLOCK;             // 512

    hipLaunchKernelGGL(gated_pool_wmma_kernel, dim3(blocks), dim3(NTHREADS), 0, stream,
                       X, G, Gb, out);
}